// CrossCompressUnit_41180146434150
// MI455X (gfx1250) — compile-verified
//
#include <hip/hip_runtime.h>

typedef float v2f __attribute__((ext_vector_type(2)));
typedef float v4f __attribute__((ext_vector_type(4)));
typedef float v8f __attribute__((ext_vector_type(8)));
typedef unsigned int v4u __attribute__((ext_vector_type(4)));
typedef int v4i __attribute__((ext_vector_type(4)));
typedef int v8i __attribute__((ext_vector_type(8)));

#define DDIM     128   // feature dim
#define ROWS     128   // rows per workgroup
#define TPB      256   // 8 waves (wave32)
#define LSTRIDE  140   // padded LDS row stride (floats): 16B-aligned float4s,
                       // 12*r mod 64 is a permutation -> conflict-free A reads
#define SWSTRIDE 132   // weight-table stride: 4*n mod 64 distinct -> no conflicts

// LDS layout (floats):
#define SV_OFF   0                          // sv   [128][140]
#define SE_OFF   (ROWS * LSTRIDE)           // se   [128][140]
#define SW_OFF   (2 * ROWS * LSTRIDE)       // sw   [4][132]  0=w_vv 1=w_ev 2=w_ve 3=w_ee
#define SZ_OFF   (SW_OFF + 4 * SWSTRIDE)    // szero[144]     zero page for B lanes n>=4
#define SBV_OFF  (SZ_OFF + 144)             // sbv  [128]
#define SBE_OFF  (SBV_OFF + DDIM)           // sbe  [128]
#define SD_OFF   (SBE_OFF + DDIM)           // sdots[128][4]
#define SMEM_FLOATS (SD_OFF + 4 * ROWS)

#define HAVE_TDM __has_builtin(__builtin_amdgcn_tensor_load_to_lds)

__global__ __launch_bounds__(TPB)
void cross_compress_kernel(const float* __restrict__ v,
                           const float* __restrict__ e,
                           const float* __restrict__ w_vv,
                           const float* __restrict__ w_ev,
                           const float* __restrict__ w_ve,
                           const float* __restrict__ w_ee,
                           const float* __restrict__ b_v,
                           const float* __restrict__ b_e,
                           float* __restrict__ v_out,
                           float* __restrict__ e_out)
{
    extern __shared__ float smem[];
    float* sv    = smem + SV_OFF;
    float* se    = smem + SE_OFF;
    float* sw    = smem + SW_OFF;
    float* szero = smem + SZ_OFF;
    float* sbv   = smem + SBV_OFF;
    float* sbe   = smem + SBE_OFF;
    float* sdots = smem + SD_OFF;

    const int tid = threadIdx.x;
    const int wave = tid >> 5;
    const int lane = tid & 31;
    const size_t rowbase = (size_t)blockIdx.x * ROWS;

    // ---------- Phase 0: stage v/e tiles (TDM) + params into LDS ----------
#if HAVE_TDM
    {
        // Each wave DMAs one 32x128 f32 slab: waves 0-3 -> v, waves 4-7 -> e.
        // TDM padding (interval 128 DWORDs, amount 12 DWORDs) realizes the
        // 140-float LDS row stride in hardware.
        const int q = wave & 3;
        const float* src = (wave < 4) ? v : e;
        unsigned long long ga =
            (unsigned long long)(const void*)(src + (rowbase + (size_t)q * 32) * DDIM);
        unsigned lds_bytes =
            ((wave < 4 ? SV_OFF : SE_OFF) + q * 32 * LSTRIDE) * sizeof(float);

        v4u g0;
        g0.x = 1u;                                    // count=1 (valid user D#)
        g0.y = lds_bytes;                             // lds_addr [63:32]
        g0.z = (unsigned)(ga & 0xffffffffu);          // global_addr lo
        g0.w = (unsigned)(ga >> 32) | (2u << 30);     // global_addr hi | type=2

        v8i g1;
        g1[0] = (2 << 16)      // data_size = 4 bytes
              | (1 << 20)      // pad_enable
              | (6 << 22)      // pad_interval: 2^(6+1) = 128 DWORDs
              | (11 << 25);    // pad_amount: 11+1 = 12 DWORDs
        g1[1] = (DDIM << 16);  // tensor_dim0[15:0] in bits[63:48]
        g1[2] = (32   << 16);  // tensor_dim1[15:0] in bits[95:80]
        g1[3] = (DDIM << 16);  // tile_dim0 in bits[127:112]
        g1[4] = 32;            // tile_dim1 in bits[143:128]
        g1[5] = DDIM;          // tensor_dim0_stride (elements)
        g1[6] = 0;
        g1[7] = 0;

        v4i z4 = {0, 0, 0, 0};
#if __has_include(<hip/amd_detail/amd_gfx1250_TDM.h>)
        v8i z8 = {0, 0, 0, 0, 0, 0, 0, 0};
        __builtin_amdgcn_tensor_load_to_lds(g0, g1, z4, z4, z8, 0);   // clang-23 form
#else
        __builtin_amdgcn_tensor_load_to_lds(g0, g1, z4, z4, 0);       // ROCm 7.2 form
#endif
    }
#else
    {
        const v4f* vg = (const v4f*)(v + rowbase * DDIM);
        const v4f* eg = (const v4f*)(e + rowbase * DDIM);
        #pragma unroll
        for (int i = 0; i < (ROWS * DDIM / 4) / TPB; ++i) {
            int f   = tid + i * TPB;
            int row = f >> 5;
            int c   = (f & 31) << 2;
            v4f pv = vg[f];
            v4f pe = eg[f];
            *(v4f*)&sv[row * LSTRIDE + c] = pv;
            *(v4f*)&se[row * LSTRIDE + c] = pe;
        }
    }
#endif
    if (tid < DDIM) {
        sw[0 * SWSTRIDE + tid] = w_vv[tid];
        sw[1 * SWSTRIDE + tid] = w_ev[tid];
        sw[2 * SWSTRIDE + tid] = w_ve[tid];
        sw[3 * SWSTRIDE + tid] = w_ee[tid];
        sbv[tid] = b_v[tid];
        sbe[tid] = b_e[tid];
    }
    if (tid < 144) szero[tid] = 0.0f;

#if HAVE_TDM
#if __has_builtin(__builtin_amdgcn_s_wait_tensorcnt)
    __builtin_amdgcn_s_wait_tensorcnt(0);
#else
    asm volatile("s_wait_tensorcnt 0x0" ::: "memory");
#endif
#endif
    __syncthreads();

    // ---------- Phase 1: per-wave WMMA dots: [16x128] @ [128x4] ----------
    {
        const int lrow = lane & 15;   // M (for A) / N (for B)
        const int half = lane >> 4;   // K-half selector
        const int r0   = wave * 16;   // this wave's 16 rows

        v8f acc_v = {};   // V_tile @ [w_vv w_ev w_ve w_ee]
        v8f acc_e = {};   // E_tile @ [w_vv w_ev w_ve w_ee]

        // A (f32 16x4): lane holds M=lrow; VGPR0=K(2*half), VGPR1=K(2*half+1)
        const float* av = &sv[(r0 + lrow) * LSTRIDE + 2 * half];
        const float* ae = &se[(r0 + lrow) * LSTRIDE + 2 * half];
        // B (f32 4x16): lane holds N=lrow; lanes n>=4 read the LDS zero page
        // (same-address broadcast) -> no EXEC divergence at the WMMA site.
        const float* bw = (lrow < 4) ? &sw[lrow * SWSTRIDE + 2 * half]
                                     : &szero[2 * half];

        #pragma unroll
        for (int t = 0; t < 32; ++t) {   // K = 4*t .. 4*t+3
            v2f a_v = *(const v2f*)(av + 4 * t);
            v2f a_e = *(const v2f*)(ae + 4 * t);
            v2f bb  = *(const v2f*)(bw + 4 * t);
            acc_v = __builtin_amdgcn_wmma_f32_16x16x4_f32(
                false, a_v, false, bb, (short)0, acc_v, false, false);
            acc_e = __builtin_amdgcn_wmma_f32_16x16x4_f32(
                false, a_e, false, bb, (short)0, acc_e, false, false);
        }

        // D: VGPR j, lanes 0-15 -> row r0+j, lanes 16-31 -> row r0+8+j; lane = N.
        // sdots[row] = {e.w_vv(n=0,E), v.w_ev(n=1,V), e.w_ve(n=2,E), v.w_ee(n=3,V)}
        if (lrow < 4) {
            const v8f& src = (lrow & 1) ? acc_v : acc_e;
            #pragma unroll
            for (int j = 0; j < 8; ++j)
                sdots[(r0 + 8 * half + j) * 4 + lrow] = src[j];
        }
    }
    __syncthreads();

    // ---------- Phase 2: elementwise combine out of LDS, NT streamed stores ----------
    {
        v4f* vo = (v4f*)(v_out + rowbase * DDIM);
        v4f* eo = (v4f*)(e_out + rowbase * DDIM);
        #pragma unroll
        for (int i = 0; i < (ROWS * DDIM / 4) / TPB; ++i) {
            int f   = tid + i * TPB;
            int row = f >> 5;
            int c   = (f & 31) << 2;
            v4f pv = *(const v4f*)&sv[row * LSTRIDE + c];
            v4f pe = *(const v4f*)&se[row * LSTRIDE + c];
            float d_vv = sdots[row * 4 + 0];
            float d_ev = sdots[row * 4 + 1];
            float d_ve = sdots[row * 4 + 2];
            float d_ee = sdots[row * 4 + 3];
            v4f bv = *(const v4f*)&sbv[c];
            v4f be = *(const v4f*)&sbe[c];
            v4f ov = pv * d_vv + pe * d_ev + bv;
            v4f oe = pv * d_ve + pe * d_ee + be;
            __builtin_nontemporal_store(ov, vo + f);
            __builtin_nontemporal_store(oe, eo + f);
        }
    }
}

extern "C" void kernel_launch(void* const* d_in, const int* in_sizes, int n_in,
                              void* d_out, int out_size, void* d_ws, size_t ws_size,
                              hipStream_t stream) {
    const float* v    = (const float*)d_in[0];
    const float* e    = (const float*)d_in[1];
    const float* w_vv = (const float*)d_in[2];
    const float* w_ev = (const float*)d_in[3];
    const float* w_ve = (const float*)d_in[4];
    const float* w_ee = (const float*)d_in[5];
    const float* b_v  = (const float*)d_in[6];
    const float* b_e  = (const float*)d_in[7];

    const int Btotal = in_sizes[0] / DDIM;          // 131072
    float* v_out = (float*)d_out;                   // [B, D]
    float* e_out = v_out + (size_t)Btotal * DDIM;   // [B, D]

    const int grid = Btotal / ROWS;                 // 1024
    const size_t smem_bytes = SMEM_FLOATS * sizeof(float);  // ~146 KB

    cross_compress_kernel<<<grid, TPB, smem_bytes, stream>>>(
        v, e, w_vv, w_ev, w_ve, w_ee, b_v, b_e, v_out, e_out);
}